// S4Kernel_63599875719315
// MI455X (gfx1250) — compile-verified
//
#include <hip/hip_runtime.h>
#include <math.h>

typedef __attribute__((ext_vector_type(2))) float v2f;
typedef __attribute__((ext_vector_type(8))) float v8f;

#define D_MODEL 1024
#define D_STATE 64
#define SEQ_L   2048
#define PI_F    3.14159265358979323846f
#define LT      2            // l-tiles (of 16 columns) per wave -> 32 l per wave

// ---------------- pre-pass: bilinear-discretization coefficients -------------
// For each (d,n), with z_inv = c + i*s on the unit circle (c^2+s^2 == 1):
//   Re(num*conj(denom)) = u0 + u1*c + u2*s
//   Im(num*conj(denom)) = v0 + v1*c + v2*s
//   |denom|^2           = k0 + k1*c + k2*s
// H[d,l] = sum_n (Re + i*Im) * rcp(Den); the rank-3 maps are exact 16x16x4
// fp32 WMMA matmuls over (n,l) tiles.
//
// Coefficients are stored directly in the WMMA A-operand per-lane layout:
// for each of the 3 matrices, two float2 K-pairs: half 0 = (K0,K1) for lanes
// 0-15, half 1 = (K2, 0) for lanes 16-31. Lanes load their pair with a single
// global_load_b64 — no cndmask shuffling in the hot loop.
__global__ void s4_coeff_kernel(const float* __restrict__ log_dt,
                                const float* __restrict__ A_real,
                                const float* __restrict__ A_imag,
                                const float* __restrict__ C,
                                float2* __restrict__ cf) {
  int idx = blockIdx.x * blockDim.x + threadIdx.x;   // d*64 + n
  if (idx >= D_MODEL * D_STATE) return;
  int d = idx >> 6;
  float dt = expf(log_dt[d]);
  float xr = 0.5f * dt * A_real[idx];                // x = dt*A/2
  float xi = 0.5f * dt * A_imag[idx];
  float dr = 1.0f - xr, di = -xi;                    // 1 - x
  float inv = 1.0f / (dr * dr + di * di);
  float a  = ((1.0f + xr) * dr + xi * di) * inv;     // Re(A_bar)
  float b  = (xi * dr - (1.0f + xr) * di) * inv;     // Im(A_bar)
  float br =  dt * dr * inv;                         // Re(B_bar)
  float bi = -dt * di * inv;                         // Im(B_bar)
  float cr = C[2 * idx], ci = C[2 * idx + 1];
  float p = cr * br - ci * bi;                       // Re(C * B_bar)
  float q = cr * bi + ci * br;                       // Im(C * B_bar)
  int o = idx * 6;
  cf[o + 0] = make_float2(p, -(p * a + q * b));      // u: (K0,K1)
  cf[o + 1] = make_float2(-(q * a - p * b), 0.0f);   // u: (K2,K3)
  cf[o + 2] = make_float2(q, -(q * a + p * b));      // v: (K0,K1)
  cf[o + 3] = make_float2(q * b - p * a, 0.0f);      // v: (K2,K3)
  cf[o + 4] = make_float2(1.0f + a * a + b * b, -2.0f * a);  // k: (K0,K1)
  cf[o + 5] = make_float2(2.0f * b, 0.0f);           // k: (K2,K3)
}

// ---------------- main pass: WMMA Cauchy reduction ---------------------------
// One wave32 per (d, 32-wide l block). Per n-tile (16 states) the three A
// operands are loaded once and feed 2 column tiles (2x3 WMMAs); epilogue per
// tile is 8 v_rcp_f32 + 8 fma per lane (packs to v_pk_fma_f32). Row sums fold
// per-lane, then across lane halves with shfl_xor(16) per the 16x16 f32 C/D
// layout. __launch_bounds__(.,4) caps the register budget so the scheduler
// does not balloon past 256 VGPRs (round-2 issue: s_set_vgpr_msb, ~1-3
// waves/SIMD occupancy).
__global__ void __launch_bounds__(256, 4)
s4_cauchy_wmma_kernel(const v2f* __restrict__ cf, float2* __restrict__ out) {
  const int lane  = threadIdx.x & 31;
  const int wave  = threadIdx.x >> 5;
  const int tile  = blockIdx.x * 8 + wave;           // 65536 super-tiles
  const int d     = tile >> 6;                       // 64 super-tiles per d
  const int lbase = (tile & 63) << 5;                // 32 l-values per wave
  const int col   = lane & 15;
  const bool hi   = lane >= 16;

  // B (4x16, KxN): V0 = rows {K0 | K2} across lane halves, V1 = {K1 | K3}.
  // Rows: K0 = 1, K1 = c, K2 = s, K3 = 0 (pad).
  v2f   B[LT];
  float accRe[LT], accIm[LT];
#pragma unroll
  for (int t = 0; t < LT; ++t) {
    float sn, cs;
    __sincosf(-2.0f * PI_F * (float)(lbase + 16 * t + col) * (1.0f / (float)SEQ_L),
              &sn, &cs);
    B[t].x = hi ? sn : 1.0f;
    B[t].y = hi ? 0.0f : cs;
    accRe[t] = 0.0f;
    accIm[t] = 0.0f;
  }

  // A (16x4, MxK): lane's K-pair preselected by layout; +hi picks (K2,K3).
  const v2f* base = cf + (size_t)(d * D_STATE + col) * 6 + (hi ? 1 : 0);

#pragma unroll
  for (int nt = 0; nt < 4; ++nt) {
    v2f Au = base[nt * 96 + 0];
    v2f Av = base[nt * 96 + 2];
    v2f Ak = base[nt * 96 + 4];
    v8f z0 = {};
#pragma unroll
    for (int t = 0; t < LT; ++t) {
      v8f Re  = __builtin_amdgcn_wmma_f32_16x16x4_f32(false, Au, false, B[t],
                                                      (short)0, z0, false, false);
      v8f Im  = __builtin_amdgcn_wmma_f32_16x16x4_f32(false, Av, false, B[t],
                                                      (short)0, z0, false, false);
      v8f Den = __builtin_amdgcn_wmma_f32_16x16x4_f32(false, Ak, false, B[t],
                                                      (short)0, z0, false, false);
#pragma unroll
      for (int m = 0; m < 8; ++m) {
        float r = __builtin_amdgcn_rcpf(Den[m]);     // v_rcp_f32 (TRANS pipe)
        accRe[t] = fmaf(Re[m], r, accRe[t]);
        accIm[t] = fmaf(Im[m], r, accIm[t]);
      }
    }
  }

  // Lane L (N=L) and lane L+16 (same column, rows M=8..15) hold half-sums.
#pragma unroll
  for (int t = 0; t < LT; ++t) {
    float re = accRe[t] + __shfl_xor(accRe[t], 16, 32);
    float im = accIm[t] + __shfl_xor(accIm[t], 16, 32);
    if (lane < 16)
      out[(size_t)d * SEQ_L + lbase + 16 * t + col] = make_float2(re, im);
  }
}

extern "C" void kernel_launch(void* const* d_in, const int* in_sizes, int n_in,
                              void* d_out, int out_size, void* d_ws, size_t ws_size,
                              hipStream_t stream) {
  const float* log_dt = (const float*)d_in[0];
  const float* A_real = (const float*)d_in[1];
  const float* A_imag = (const float*)d_in[2];
  const float* C      = (const float*)d_in[3];
  // d_in[4] = L (python scalar) — fixed at compile time (2048).

  float2* cf = (float2*)d_ws;  // 1024*64*6 float2 = 3 MB scratch

  s4_coeff_kernel<<<(D_MODEL * D_STATE + 255) / 256, 256, 0, stream>>>(
      log_dt, A_real, A_imag, C, cf);

  const int tiles = D_MODEL * (SEQ_L / (16 * LT));   // 65536 waves
  s4_cauchy_wmma_kernel<<<tiles / 8, 256, 0, stream>>>(
      (const v2f*)cf, (float2*)d_out);
}